// LogitFit_20117626815073
// MI455X (gfx1250) — compile-verified
//
#include <hip/hip_runtime.h>

typedef __attribute__((ext_vector_type(2))) float v2f;
typedef __attribute__((ext_vector_type(8))) float v8f;

#define BATCHES 8
#define HW      307200
#define TILES   4800      // HW / 64
#define NWG     120       // workgroups per batch for big passes
#define ACC_STRIDE 4224   // per-batch accumulator: 4096 XtX + 64 Xty + N + Syy + pad

__device__ __forceinline__ v8f wmma_f32(v2f a, v2f b, v8f c) {
  // D = A(16x4) * B(4x16) + C, fp32
  return __builtin_amdgcn_wmma_f32_16x16x4_f32(false, a, false, b, (short)0, c, false, false);
}

// ---------------------------------------------------------------------------
// Pass 1: accumulate XtX (64x64), Xty (64), N, Syy per batch over masked pixels
// ---------------------------------------------------------------------------
__global__ void __launch_bounds__(256) xtx_kernel(const float* __restrict__ bases,
                                                  const float* __restrict__ targets,
                                                  float* __restrict__ acc) {
  const int b   = blockIdx.x / NWG;
  const int wgi = blockIdx.x % NWG;
  const float* Xb = bases   + (size_t)b * 63 * HW;
  const float* Yb = targets + (size_t)b * HW;
  float* accb = acc + b * ACC_STRIDE;

  __shared__ float Xs[64][66];   // masked basis tile [channel][pixel]
  __shared__ float Ys[64];       // masked target tile

  const int t    = threadIdx.x;
  const int lane = t & 31;
  const int wv   = t >> 5;             // wave 0..7
  const int rb0  = wv >> 2;            // row-blocks 0..1
  const int cb   = wv & 3;             // col-block 0..3
  const int rb1  = rb0 + 2;
  const int li   = lane & 15;
  const int koff = (lane >> 4) << 1;   // 0 or 2

  v8f c0 = {}; v8f c1 = {};
  float xtyAcc = 0.f, nAcc = 0.f, syyAcc = 0.f;
  const int xc  = t & 63;              // channel this thread accumulates Xty for
  const int xps = (t >> 6) * 16;       // its 16-pixel sub-range

  for (int tile = wgi; tile < TILES; tile += NWG) {
    const int p0 = tile * 64;
    if (t < 64) {
      float y  = Yb[p0 + t];
      float mf = (__builtin_fabsf(y) < __builtin_inff()) ? 1.0f : 0.0f;
      float yf = (mf != 0.0f) ? y : 0.0f;
      Ys[t]    = yf;
      Xs[0][t] = mf;                   // homogeneous ones channel, masked
      nAcc   += mf;
      syyAcc += yf * yf;
    }
    __syncthreads();
    // fill channels 1..63, masked (coalesced 256B rows)
    for (int idx = t; idx < 63 * 64; idx += 256) {
      int c = idx >> 6, p = idx & 63;
      Xs[c + 1][p] = Xb[(size_t)c * HW + p0 + p] * Xs[0][p];
    }
    __syncthreads();

    // Xty partial (VALU, tiny)
    {
      const float* row = &Xs[xc][0];
      float s = 0.f;
      #pragma unroll
      for (int p = 0; p < 16; ++p) s += row[xps + p] * Ys[xps + p];
      xtyAcc += s;
    }

    // XtX: 16 K-steps of 4 pixels, 2 output tiles per wave
    #pragma unroll
    for (int k0 = 0; k0 < 64; k0 += 4) {
      v2f a0 = *(const v2f*)&Xs[rb0 * 16 + li][k0 + koff];
      v2f a1 = *(const v2f*)&Xs[rb1 * 16 + li][k0 + koff];
      v2f bb = *(const v2f*)&Xs[cb  * 16 + li][k0 + koff];
      c0 = wmma_f32(a0, bb, c0);
      c1 = wmma_f32(a1, bb, c1);
    }
    __syncthreads();   // before next tile overwrites LDS
  }

  // flush accumulators
  const int mhi = (lane >> 4) * 8;
  #pragma unroll
  for (int r = 0; r < 8; ++r) {
    int col  = cb * 16 + li;
    atomicAdd(&accb[(rb0 * 16 + r + mhi) * 64 + col], c0[r]);
    atomicAdd(&accb[(rb1 * 16 + r + mhi) * 64 + col], c1[r]);
  }
  atomicAdd(&accb[4096 + xc], xtyAcc);
  if (t < 64) {
    atomicAdd(&accb[4160], nAcc);
    atomicAdd(&accb[4161], syyAcc);
  }
}

// ---------------------------------------------------------------------------
// Pass 2: per batch — Cholesky, solve w, scalar beta fixed-point, inverse/beta
// ---------------------------------------------------------------------------
__global__ void __launch_bounds__(64) solve_kernel(const float* __restrict__ acc,
                                                   float* __restrict__ Ms,
                                                   float* __restrict__ wWs,
                                                   float* __restrict__ outW) {
  const int b = blockIdx.x;
  const int t = threadIdx.x;     // 0..63
  __shared__ float A[64][65];
  __shared__ float xty[64];
  __shared__ float wv[64];
  __shared__ float sc[1];

  const float* accb = acc + b * ACC_STRIDE;
  for (int i = t; i < 4096; i += 64) A[i >> 6][i & 63] = accb[i];
  xty[t] = accb[4096 + t];
  __syncthreads();
  const float Nv  = accb[4160];
  const float Syy = accb[4161];

  // Cholesky A = L L^T (lower)
  for (int k = 0; k < 64; ++k) {
    if (t == 0) A[k][k] = sqrtf(A[k][k]);
    __syncthreads();
    if (t > k) A[t][k] *= (1.0f / A[k][k]);
    __syncthreads();
    if (t > k) {
      float lik = A[t][k];
      for (int j = k + 1; j <= t; ++j) A[t][j] -= lik * A[j][k];
    }
    __syncthreads();
  }

  if (t == 0) {
    // solve XtX w = Xty  (beta scaling cancels)
    float y[64];
    for (int i = 0; i < 64; ++i) {
      float s = xty[i];
      for (int j = 0; j < i; ++j) s -= A[i][j] * y[j];
      y[i] = s / A[i][i];
    }
    for (int i = 63; i >= 0; --i) {
      float s = y[i];
      for (int j = i + 1; j < 64; ++j) s -= A[j][i] * wv[j];
      wv[i] = s / A[i][i];
    }
    // E_d = Syy - w.Xty (exact-solve identity), then beta fixed point
    float ed = Syy;
    for (int i = 0; i < 64; ++i) ed -= wv[i] * xty[i];
    float beta0 = sqrtf(Nv), beta = beta0;
    bool done = false;
    #pragma unroll
    for (int it = 0; it < 5; ++it) {
      float bn   = Nv / (ed + 64.0f / beta);
      bool  conv = (fabsf(bn / beta0 - 1.0f) < 0.02f);
      if (!done) { beta = bn; beta0 = bn; }
      done = done || conv;
    }
    sc[0] = beta;
  }
  __syncthreads();

  // column t of XtX^{-1}, scaled by 1/beta  -> Ms
  const float invb = 1.0f / sc[0];
  float x[64];
  for (int i = 0; i < 64; ++i) {
    float s = (i == t) ? 1.0f : 0.0f;
    for (int j = 0; j < i; ++j) s -= A[i][j] * x[j];
    x[i] = s / A[i][i];
  }
  for (int i = 63; i >= 0; --i) {
    float s = x[i];
    for (int j = i + 1; j < 64; ++j) s -= A[j][i] * x[j];
    x[i] = s / A[i][i];
  }
  float* Msb = Ms + b * 4096;
  for (int i = 0; i < 64; ++i) Msb[i * 64 + t] = x[i] * invb;
  wWs[b * 64 + t]  = wv[t];
  outW[b * 64 + t] = wv[t];
}

// ---------------------------------------------------------------------------
// Pass 3: logits = w.b, var = b^T Ms b per pixel (Ms A-frags live in VGPRs)
// ---------------------------------------------------------------------------
__global__ void __launch_bounds__(256) var_kernel(const float* __restrict__ bases,
                                                  const float* __restrict__ Ms,
                                                  const float* __restrict__ wWs,
                                                  float* __restrict__ outLogit,
                                                  float* __restrict__ outVar) {
  const int b   = blockIdx.x / NWG;
  const int wgi = blockIdx.x % NWG;
  const float* Xb  = bases + (size_t)b * 63 * HW;
  const float* Msb = Ms + b * 4096;

  __shared__ float Bs[64][66];
  __shared__ float varS[64];
  __shared__ float wS[64];

  const int t    = threadIdx.x;
  const int lane = t & 31;
  const int wv   = t >> 5;
  const int rb0  = wv >> 2, cb = wv & 3, rb1 = rb0 + 2;
  const int li   = lane & 15;
  const int koff = (lane >> 4) << 1;
  const int mhi  = (lane >> 4) * 8;

  if (t < 64) wS[t] = wWs[b * 64 + t];

  // Preload Ms A-fragments (constant over all pixel tiles): 64 VGPRs
  v2f mA0[16], mA1[16];
  #pragma unroll
  for (int ks = 0; ks < 16; ++ks) {
    int k = ks * 4 + koff;
    mA0[ks] = *(const v2f*)&Msb[(rb0 * 16 + li) * 64 + k];
    mA1[ks] = *(const v2f*)&Msb[(rb1 * 16 + li) * 64 + k];
  }

  for (int tile = wgi; tile < TILES; tile += NWG) {
    const int p0 = tile * 64;
    if (t < 64) { Bs[0][t] = 1.0f; varS[t] = 0.0f; }
    for (int idx = t; idx < 63 * 64; idx += 256) {
      int c = idx >> 6, p = idx & 63;
      Bs[c + 1][p] = Xb[(size_t)c * HW + p0 + p];
    }
    __syncthreads();

    // T = Ms @ Btile via WMMA (2 tiles per wave, K=64)
    v8f c0 = {}, c1 = {};
    #pragma unroll
    for (int ks = 0; ks < 16; ++ks) {
      int k = ks * 4 + koff;
      v2f bb;
      bb.x = Bs[k][cb * 16 + li];
      bb.y = Bs[k + 1][cb * 16 + li];
      c0 = wmma_f32(mA0[ks], bb, c0);
      c1 = wmma_f32(mA1[ks], bb, c1);
    }

    // var[p] += sum_m T[m][p] * B[m][p]
    const int pcol = cb * 16 + li;
    float s0 = 0.f;
    #pragma unroll
    for (int r = 0; r < 8; ++r) {
      s0 += c0[r] * Bs[rb0 * 16 + r + mhi][pcol];
      s0 += c1[r] * Bs[rb1 * 16 + r + mhi][pcol];
    }
    atomicAdd(&varS[pcol], s0);
    __syncthreads();

    if (t < 64) {
      float lg = 0.f;
      #pragma unroll
      for (int c = 0; c < 64; ++c) lg += wS[c] * Bs[c][t];
      outLogit[(size_t)b * HW + p0 + t] = lg;
      outVar  [(size_t)b * HW + p0 + t] = varS[t];
    }
    __syncthreads();   // before refill
  }
}

// ---------------------------------------------------------------------------
extern "C" void kernel_launch(void* const* d_in, const int* in_sizes, int n_in,
                              void* d_out, int out_size, void* d_ws, size_t ws_size,
                              hipStream_t stream) {
  const float* bases   = (const float*)d_in[0];   // (8, 63, 480, 640)
  const float* targets = (const float*)d_in[1];   // (8, 1, 480, 640)

  float* out      = (float*)d_out;
  float* outLogit = out;                                  // 8*HW
  float* outW     = out + (size_t)BATCHES * HW;           // 8*64
  float* outVar   = outW + BATCHES * 64;                  // 8*HW

  float* ws  = (float*)d_ws;
  float* acc = ws;                                        // 8*4224
  float* Msv = ws + BATCHES * ACC_STRIDE;                 // 8*4096
  float* wWs = Msv + BATCHES * 4096;                      // 8*64

  hipMemsetAsync(acc, 0, (size_t)BATCHES * ACC_STRIDE * sizeof(float), stream);

  xtx_kernel <<<dim3(BATCHES * NWG), 256, 0, stream>>>(bases, targets, acc);
  solve_kernel<<<dim3(BATCHES),       64, 0, stream>>>(acc, Msv, wWs, outW);
  var_kernel <<<dim3(BATCHES * NWG), 256, 0, stream>>>(bases, Msv, wWs, outLogit, outVar);
}